// CameraOptimizer_90812788506815
// MI455X (gfx1250) — compile-verified
//
#include <hip/hip_runtime.h>
#include <hip/hip_bf16.h>
#include <math.h>

typedef __attribute__((ext_vector_type(2))) float v2f;
typedef __attribute__((ext_vector_type(8))) float v8f;
typedef __attribute__((address_space(3))) char lds_char;

#define NCAM 1000
#define CAM_STRIDE 16   // floats per camera: R(9) | t(3) | K(4)  = 64 bytes

// ---------------------------------------------------------------------------
// Kernel 1: per-camera prep. One wave handles 5 cameras via one
// V_WMMA_F32_16X16X4_F32 (block-diagonal packing of 3x3 matmuls).
// ---------------------------------------------------------------------------
__global__ __launch_bounds__(256) void cam_prep_kernel(
    const float* __restrict__ noisy_K,
    const float* __restrict__ noisy_R,
    const float* __restrict__ noisy_t,
    const float* __restrict__ intr_d,
    const float* __restrict__ rot_d,
    const float* __restrict__ trans_d,
    float* __restrict__ cam_tab)
{
    const int lane = threadIdx.x & 31;
    const int wave = (blockIdx.x * blockDim.x + threadIdx.x) >> 5;  // 0..199
    const int cb   = wave * 5;                                      // camera base
    const int half = lane >> 4;     // 0: K=0,1  1: K=2,3 (A); K-rows for B
    const int m    = lane & 15;     // A row index / B col index
    const int cA   = m / 3;         // camera-in-group (0..4; 5 for pad row 15)
    const int i    = m - 3 * cA;    // row (A) / col (B) within 3x3
    const int camA = cb + (cA < 5 ? cA : 0);

    // --- Rodrigues for rotation_deltas[camA] (each lane computes its camera) ---
    float wx = rot_d[3*camA+0], wy = rot_d[3*camA+1], wz = rot_d[3*camA+2];
    float th  = sqrtf(wx*wx + wy*wy + wz*wz);
    float inv = 1.0f / fmaxf(th, 1e-8f);
    float kx = wx*inv, ky = wy*inv, kz = wz*inv;
    float sn = sinf(th), cs = cosf(th);
    float Km[3][3] = {{0.f,-kz, ky},{ kz,0.f,-kx},{-ky, kx,0.f}};
    float Rd[3][3];
    #pragma unroll
    for (int r = 0; r < 3; ++r) {
        #pragma unroll
        for (int c = 0; c < 3; ++c) {
            float k2 = Km[r][0]*Km[0][c] + Km[r][1]*Km[1][c] + Km[r][2]*Km[2][c];
            Rd[r][c] = ((r == c) ? 1.f : 0.f) + sn*Km[r][c] + (1.f - cs)*k2;
        }
    }

    // --- A fragment: 16x4 f32. lanes 0-15 -> K0,K1 ; lanes 16-31 -> K2,K3 ---
    v2f a;
    if (half == 0) { a.x = Rd[i][0]; a.y = Rd[i][1]; }
    else           { a.x = Rd[i][2]; a.y = 0.f;      }
    if (m == 15)   { a.x = 0.f;      a.y = 0.f;      }   // pad row

    // --- B fragment: 4x16 f32, col m holds noisy_R[:, j] of camera cA ---
    const float* nR = noisy_R + 9*camA;
    v2f b;
    if (half == 0) { b.x = nR[0*3+i]; b.y = nR[1*3+i]; }
    else           { b.x = nR[2*3+i]; b.y = 0.f;       }
    if (m == 15)   { b.x = 0.f;       b.y = 0.f;       } // pad col

    v8f acc = {};
    acc = __builtin_amdgcn_wmma_f32_16x16x4_f32(
        /*neg_a=*/false, a, /*neg_b=*/false, b,
        /*c_mod=*/(short)0, acc, /*reuse_a=*/false, /*reuse_b=*/false);

    // --- extract diagonal 3x3 blocks: C layout lane=N, VGPR r -> M=r+8*half ---
    #pragma unroll
    for (int r = 0; r < 8; ++r) {
        int M  = r + 8*half;
        int c  = M / 3;
        int ii = M - 3*c;
        int jj = m - 3*c;
        if (M < 15 && m < 15 && jj >= 0 && jj < 3)
            cam_tab[(cb + c)*CAM_STRIDE + 3*ii + jj] = acc[r];
    }

    // --- t_corr and K_corr: one lane per camera in this wave's group ---
    if (lane < 5) {
        int c = cb + lane;
        float* dst = cam_tab + c*CAM_STRIDE;
        dst[ 9] = noisy_t[3*c+0] + trans_d[3*c+0];
        dst[10] = noisy_t[3*c+1] + trans_d[3*c+1];
        dst[11] = noisy_t[3*c+2] + trans_d[3*c+2];
        dst[12] = noisy_K[4*c+0] + intr_d[4*c+0];
        dst[13] = noisy_K[4*c+1] + intr_d[4*c+1];
        dst[14] = noisy_K[4*c+2] + intr_d[4*c+2];
        dst[15] = noisy_K[4*c+3] + intr_d[4*c+3];
    }
}

// ---------------------------------------------------------------------------
// Kernel 2: per-point projection. Stage the 64 KB camera table into LDS with
// CDNA5 async global->LDS b128 transfers, then gather from LDS per point.
// ---------------------------------------------------------------------------
__global__ __launch_bounds__(256) void project_kernel(
    const float* __restrict__ Xw,
    const int*   __restrict__ cidx,
    const float* __restrict__ cam_tab,
    float*       __restrict__ out, int N)
{
    __shared__ float s_cam[NCAM * CAM_STRIDE];   // 64000 bytes

    // async stage: 4000 x 16-byte chunks, per-lane LDS-direct transfers
    unsigned ldsBase = (unsigned)(unsigned long long)(lds_char*)s_cam;
    const int nChunks = (NCAM * CAM_STRIDE * 4) / 16;   // 4000
    for (int ch = threadIdx.x; ch < nChunks; ch += blockDim.x) {
        unsigned goff  = (unsigned)ch * 16u;
        unsigned laddr = ldsBase + goff;
        asm volatile("global_load_async_to_lds_b128 %0, %1, %2 offset:0"
                     :: "v"(laddr), "v"(goff), "s"(cam_tab) : "memory");
    }
    asm volatile("s_wait_asynccnt 0" ::: "memory");
    __syncthreads();

    const int stride = gridDim.x * blockDim.x;
    for (int p = blockIdx.x * blockDim.x + threadIdx.x; p < N; p += stride) {
        int c = cidx[p];
        const float4* cp = (const float4*)(s_cam + c * CAM_STRIDE);
        float4 q0 = cp[0];  // R00 R01 R02 R10
        float4 q1 = cp[1];  // R11 R12 R20 R21
        float4 q2 = cp[2];  // R22 t0  t1  t2
        float4 q3 = cp[3];  // K0  K1  K2  K3

        float x = Xw[3*p+0], y = Xw[3*p+1], z = Xw[3*p+2];
        float X0 = q0.x*x + q0.y*y + q0.z*z + q2.y;
        float X1 = q0.w*x + q1.x*y + q1.y*z + q2.z;
        float X2 = q1.z*x + q1.w*y + q2.x*z + q2.w;
        float invz = 1.0f / X2;
        float u = q3.x * X0 * invz + q3.z;
        float v = q3.y * X1 * invz + q3.w;
        float2 r; r.x = u; r.y = v;
        *(float2*)(out + 2*p) = r;
    }
}

extern "C" void kernel_launch(void* const* d_in, const int* in_sizes, int n_in,
                              void* d_out, int out_size, void* d_ws, size_t ws_size,
                              hipStream_t stream) {
    const float* Xw  = (const float*)d_in[0];
    const int*   ci  = (const int*)  d_in[1];
    const float* nK  = (const float*)d_in[2];
    const float* nR  = (const float*)d_in[3];
    const float* nt  = (const float*)d_in[4];
    const float* idl = (const float*)d_in[5];
    const float* rdl = (const float*)d_in[6];
    const float* tdl = (const float*)d_in[7];
    float* cam_tab = (float*)d_ws;          // 64000 bytes used
    float* out     = (float*)d_out;
    const int N = in_sizes[1];              // NUM_POINTS

    // 25 blocks * 256 threads = 200 waves -> 5 cameras each = 1000 cameras
    cam_prep_kernel<<<25, 256, 0, stream>>>(nK, nR, nt, idl, rdl, tdl, cam_tab);
    // memory-bound streaming pass; 1024 blocks, grid-stride over 4.19M points
    project_kernel<<<1024, 256, 0, stream>>>(Xw, ci, cam_tab, out, N);
}